// ImportedNetwork_14027363188907
// MI455X (gfx1250) — compile-verified
//
#include <hip/hip_runtime.h>

typedef float v2f __attribute__((ext_vector_type(2)));
typedef float v8f __attribute__((ext_vector_type(8)));

namespace {
constexpr int T  = 256;
constexpr int Bn = 1024;
constexpr int In = 12;
constexpr int Hn = 256;
constexpr int On = 7;
constexpr int BT = 16;     // batch rows per workgroup
constexpr int RSTR = 260;  // padded LDS row stride (floats): bank-conflict free

// LDS layout (float offsets)
constexpr int L_REC  = 0;                  // rec_w   256*260
constexpr int L_SPK0 = L_REC  + Hn * RSTR; // spike ping buf 16*260
constexpr int L_SPK1 = L_SPK0 + BT * RSTR; // spike pong buf 16*260
constexpr int L_FC2  = L_SPK1 + BT * RSTR; // fc2_w   7*256
constexpr int L_XT   = L_FC2  + On * Hn;   // x_t     16*12
constexpr int L_C2   = L_XT   + BT * In;   // cur2buf 128
constexpr int L_TOT  = L_C2   + 128;       // = 76992 floats = 307968 B (< 320KB)
}

__global__ __launch_bounds__(256)
void snn_scan_kernel(const float* __restrict__ x,
                     const float* __restrict__ fc1w,
                     const float* __restrict__ fc1b,
                     const float* __restrict__ recw,
                     const float* __restrict__ recb,
                     const float* __restrict__ fc2w,
                     const float* __restrict__ fc2b,
                     float* __restrict__ out2,   // [T,B,7]
                     float* __restrict__ out1)   // [T,B,256]
{
    extern __shared__ float lds[];
    const int tid = threadIdx.x;
    const int wv  = tid >> 5;   // wave id 0..7
    const int ln  = tid & 31;   // lane
    const int lh  = ln >> 4;    // lane half (0/1)
    const int lm  = ln & 15;
    const int b0  = blockIdx.x * BT;

    // ---- stage constant weights into LDS ----
    for (int idx = tid; idx < Hn * Hn; idx += 256) {
        int n = idx >> 8, k = idx & 255;
        lds[L_REC + n * RSTR + k] = recw[idx];
    }
    for (int idx = tid; idx < On * Hn; idx += 256)
        lds[L_FC2 + idx] = fc2w[idx];
    for (int idx = tid; idx < 2 * BT * RSTR; idx += 256)
        lds[L_SPK0 + idx] = 0.0f;   // spk1 initial state = 0

    // per-lane column assignment: wave wv owns N-tiles [wv*32, wv*32+32)
    const int na = wv * 32 + lm;        // tile A column
    const int nb = na + 16;             // tile B column
    float w1a[In], w1b[In];
    for (int i = 0; i < In; ++i) {
        w1a[i] = fc1w[na * In + i];
        w1b[i] = fc1w[nb * In + i];
    }
    const float bca = fc1b[na] + recb[na];
    const float bcb = fc1b[nb] + recb[nb];

    // layer-1 state in WMMA C/D layout: element r -> (M = r + 8*lh, N = col)
    v8f syn_a = {}, syn_b = {}, mem_a = {}, mem_b = {};

    // layer-2 state (owned by wave 0): lane handles pairs p = ln*4+q, m=p>>3, o=p&7
    float syn2[4] = {0.f, 0.f, 0.f, 0.f};
    float mem2[4] = {0.f, 0.f, 0.f, 0.f};
    float b2[4];
    for (int q = 0; q < 4; ++q) {
        int o = (ln * 4 + q) & 7;
        b2[q] = (o < On) ? fc2b[o] : 0.0f;
    }

    int pcur = 0;
    for (int t = 0; t < T; ++t) {
        __syncthreads();  // B0: prev iteration's consumers of x_t / cur2buf done

        if (tid < BT * In) {
            int m = tid / In, i = tid - m * In;
            lds[L_XT + tid] = x[(size_t)t * Bn * In + (size_t)(b0 + m) * In + i];
        }
        if (tid < 128) lds[L_C2 + tid] = 0.0f;

        __syncthreads();  // B1: x_t staged, cur2buf zeroed, prev spikes ready

        const float* sprev = &lds[pcur ? L_SPK1 : L_SPK0];
        float*       scur  = &lds[pcur ? L_SPK0 : L_SPK1];

        // cur1 = x_t @ fc1_w^T (+fc1_b + rec_b) into accumulator chain 0 init
        v8f ca0, cb0;
        for (int r = 0; r < 8; ++r) {
            int m = r + 8 * lh;
            float sa = bca, sb = bcb;
            for (int i = 0; i < In; ++i) {
                float xv = lds[L_XT + m * In + i];
                sa = fmaf(xv, w1a[i], sa);
                sb = fmaf(xv, w1b[i], sb);
            }
            ca0[r] = sa;
            cb0[r] = sb;
        }

        // recurrent GEMM: spk1_prev[16x256] @ rec_w^T tiles, fp32 WMMA 16x16x4.
        // K=256 split into two halves with independent accumulators:
        // 4 independent WMMA dependency chains interleaved -> better matrix-pipe ILP.
        v8f ca1 = {}, cb1 = {};
        {
            const float* arow  = sprev + lm * RSTR + 2 * lh;
            const float* browa = &lds[L_REC + na * RSTR + 2 * lh];
            const float* browb = &lds[L_REC + nb * RSTR + 2 * lh];
            constexpr int KH = Hn / 2;  // 128 (floats) second-half offset
            #pragma unroll 4
            for (int kc = 0; kc < Hn / 8; ++kc) {  // 32 iterations, 4 WMMAs each
                v2f af0 = *(const v2f*)(arow  + kc * 4);
                v2f af1 = *(const v2f*)(arow  + kc * 4 + KH);
                v2f ba0 = *(const v2f*)(browa + kc * 4);
                v2f ba1 = *(const v2f*)(browa + kc * 4 + KH);
                v2f bb0 = *(const v2f*)(browb + kc * 4);
                v2f bb1 = *(const v2f*)(browb + kc * 4 + KH);
                ca0 = __builtin_amdgcn_wmma_f32_16x16x4_f32(
                        false, af0, false, ba0, (short)0, ca0, false, false);
                cb0 = __builtin_amdgcn_wmma_f32_16x16x4_f32(
                        false, af0, false, bb0, (short)0, cb0, false, false);
                ca1 = __builtin_amdgcn_wmma_f32_16x16x4_f32(
                        false, af1, false, ba1, (short)0, ca1, false, false);
                cb1 = __builtin_amdgcn_wmma_f32_16x16x4_f32(
                        false, af1, false, bb1, (short)0, cb1, false, false);
            }
        }

        // LIF update + spike; write spikes to LDS (next step A) and global
        for (int r = 0; r < 8; ++r) {
            int m = r + 8 * lh;
            float ra = ca0[r] + ca1[r];   // merge split-K chains
            float rb = cb0[r] + cb1[r];
            float s_a = fmaf(0.9f, syn_a[r], ra);
            float s_b = fmaf(0.9f, syn_b[r], rb);
            float rsta = (mem_a[r] > 1.0f) ? 0.0f : 1.0f;  // 1-reset1
            float rstb = (mem_b[r] > 1.0f) ? 0.0f : 1.0f;
            float m_a = fmaf(0.85f, mem_a[r], s_a) * rsta;
            float m_b = fmaf(0.85f, mem_b[r], s_b) * rstb;
            float spa = (m_a > 1.0f) ? 1.0f : 0.0f;
            float spb = (m_b > 1.0f) ? 1.0f : 0.0f;
            syn_a[r] = s_a; syn_b[r] = s_b;
            mem_a[r] = m_a; mem_b[r] = m_b;
            scur[m * RSTR + na] = spa;
            scur[m * RSTR + nb] = spb;
            size_t gb = (size_t)t * Bn * Hn + (size_t)(b0 + m) * Hn;
            out1[gb + na] = spa;
            out1[gb + nb] = spb;
        }

        __syncthreads();  // B2: spk1_new complete in LDS

        // layer-2 partial GEMM: wave wv reduces K slice [32*wv, 32*wv+32)
        {
            int k0 = wv * 32;
            for (int q = 0; q < 4; ++q) {
                int p = ln * 4 + q;
                int m = p >> 3, o = p & 7;
                if (o < On) {
                    const float4* sp = (const float4*)(scur + m * RSTR + k0);
                    const float4* wp = (const float4*)(&lds[L_FC2 + o * Hn + k0]);
                    float acc = 0.0f;
                    #pragma unroll
                    for (int j = 0; j < 8; ++j) {
                        float4 s = sp[j], w = wp[j];
                        acc = fmaf(s.x, w.x, fmaf(s.y, w.y,
                              fmaf(s.z, w.z, fmaf(s.w, w.w, acc))));
                    }
                    atomicAdd(&lds[L_C2 + p], acc);  // ds_add_f32
                }
            }
        }

        __syncthreads();  // B3: cur2 reduction complete

        if (wv == 0) {
            for (int q = 0; q < 4; ++q) {
                int p = ln * 4 + q;
                int m = p >> 3, o = p & 7;
                float cur2 = lds[L_C2 + p] + b2[q];
                float rst  = (mem2[q] > 1.0f) ? 0.0f : 1.0f;  // 1-reset2
                float s2 = fmaf(0.9f, syn2[q], cur2);
                float m2 = fmaf(0.8f, mem2[q], s2) * rst;
                syn2[q] = s2; mem2[q] = m2;
                if (o < On)
                    out2[(size_t)t * Bn * On + (size_t)(b0 + m) * On + o] =
                        (m2 > 1.0f) ? 1.0f : 0.0f;
            }
        }
        pcur ^= 1;
    }
}

extern "C" void kernel_launch(void* const* d_in, const int* in_sizes, int n_in,
                              void* d_out, int out_size, void* d_ws, size_t ws_size,
                              hipStream_t stream) {
    (void)in_sizes; (void)n_in; (void)out_size; (void)d_ws; (void)ws_size;
    const float* x    = (const float*)d_in[0];
    const float* fc1w = (const float*)d_in[1];
    const float* fc1b = (const float*)d_in[2];
    const float* recw = (const float*)d_in[3];
    const float* recb = (const float*)d_in[4];
    const float* fc2w = (const float*)d_in[5];
    const float* fc2b = (const float*)d_in[6];
    float* out2 = (float*)d_out;                          // [T,B,7] first
    float* out1 = out2 + (size_t)T * Bn * On;             // then [T,B,256]

    dim3 grid(Bn / BT);   // 64 workgroups, each owns 16 batch rows
    dim3 block(256);      // 8 waves
    size_t shmem = (size_t)L_TOT * sizeof(float);         // ~301 KB of the 320 KB WGP LDS
    snn_scan_kernel<<<grid, block, shmem, stream>>>(
        x, fc1w, fc1b, recw, recb, fc2w, fc2b, out2, out1);
}